// DeepWDK_23914377904552
// MI455X (gfx1250) — compile-verified
//
#include <hip/hip_runtime.h>
#include <hip/hip_bf16.h>

#define L_SEQ 512
#define NAA 21
#define DDIM 128
#define NCOL (NAA * DDIM)   // 2688
#define NSAMP 256
#define S_ELE (NAA * NAA)   // 441
#define S_STRIDE 516        // sTile row stride (floats): 516 % 64 == 4 -> conflict-free
#define W_STRIDE 520        // wslab row stride (floats): 520 % 64 == 8 -> conflict-free

typedef float v2f __attribute__((ext_vector_type(2)));
typedef float v8f __attribute__((ext_vector_type(8)));

// ---------------------------------------------------------------------------
// Kernel 0: w = sigmoid(symmetric(w_param)), 512x512
// ---------------------------------------------------------------------------
__global__ void kernel_buildw(const float* __restrict__ wparam, float* __restrict__ w) {
    int idx = blockIdx.x * blockDim.x + threadIdx.x;
    if (idx >= L_SEQ * L_SEQ) return;
    int p = idx / L_SEQ, q = idx % L_SEQ;
    float x;
    if (p > q)      x = wparam[p * (p - 1) / 2 + q];
    else if (q > p) x = wparam[q * (q - 1) / 2 + p];
    else            x = 0.0f;
    w[idx] = 1.0f / (1.0f + expf(-x));
}

// ---------------------------------------------------------------------------
// Kernel 1: E[n, c] = b[c] + sum_l W[(l*21 + X[n,l]) * 2688 + c]
//   one-hot GEMM degenerates to a row gather-sum; n = 0..511 (X1 then X2)
// ---------------------------------------------------------------------------
__global__ __launch_bounds__(256) void kernel_network(const int* __restrict__ X1,
                                                      const int* __restrict__ X2,
                                                      const float* __restrict__ W,
                                                      const float* __restrict__ bvec,
                                                      float* __restrict__ E) {
    int n = blockIdx.x;                              // 0..511
    const int* X = (n < NSAMP) ? (X1 + (size_t)n * L_SEQ)
                               : (X2 + (size_t)(n - NSAMP) * L_SEQ);
    __shared__ int xs[L_SEQ];
    for (int t = threadIdx.x; t < L_SEQ; t += 256) xs[t] = X[t];
    __syncthreads();

    for (int g = threadIdx.x; g < NCOL / 4; g += 256) {   // 672 float4 groups
        int c = g * 4;
        float4 acc = *(const float4*)(bvec + c);
        #pragma unroll 4
        for (int l = 0; l < L_SEQ; ++l) {
            const float* row = W + ((size_t)l * NAA + xs[l]) * NCOL + c;
            float4 v = *(const float4*)row;
            acc.x += v.x; acc.y += v.y; acc.z += v.z; acc.w += v.w;
        }
        *(float4*)(E + (size_t)n * NCOL + c) = acc;
    }
}

// ---------------------------------------------------------------------------
// Kernel 2: S[n, a, b] = dot(E[n,a,:], E[n,b,:])  (21x21 Gram per sample)
// ---------------------------------------------------------------------------
__global__ __launch_bounds__(256) void kernel_gram(const float* __restrict__ E,
                                                   float* __restrict__ S) {
    int n = blockIdx.x;                              // 0..511
    __shared__ float e[NCOL];
    for (int t = threadIdx.x; t < NCOL; t += 256) e[t] = E[(size_t)n * NCOL + t];
    __syncthreads();
    for (int idx = threadIdx.x; idx < S_ELE; idx += 256) {
        int a = idx / NAA, b = idx % NAA;
        float acc = 0.0f;
        #pragma unroll 8
        for (int d = 0; d < DDIM; ++d) acc += e[a * DDIM + d] * e[b * DDIM + d];
        S[(size_t)n * S_ELE + idx] = acc;
    }
}

// ---------------------------------------------------------------------------
// Kernel 3: self kernels k[n] = d^T w d, d[l] = S[n, X[n,l], X[n,l]]
// ---------------------------------------------------------------------------
__global__ __launch_bounds__(256) void kernel_selfk(const float* __restrict__ S,
                                                    const int* __restrict__ X1,
                                                    const int* __restrict__ X2,
                                                    const float* __restrict__ w,
                                                    float* __restrict__ kk) {
    int n = blockIdx.x;                              // 0..511
    const int* X = (n < NSAMP) ? (X1 + (size_t)n * L_SEQ)
                               : (X2 + (size_t)(n - NSAMP) * L_SEQ);
    __shared__ float d[L_SEQ];
    __shared__ float red[8];
    for (int t = threadIdx.x; t < L_SEQ; t += 256) {
        int x = X[t];
        d[t] = S[(size_t)n * S_ELE + x * NAA + x];
    }
    __syncthreads();
    float acc = 0.0f;
    for (int p = threadIdx.x; p < L_SEQ; p += 256) {
        const float* wr = w + (size_t)p * L_SEQ;
        float u = 0.0f;
        #pragma unroll 8
        for (int q = 0; q < L_SEQ; ++q) u += wr[q] * d[q];
        acc += d[p] * u;
    }
    #pragma unroll
    for (int off = 16; off; off >>= 1) acc += __shfl_xor(acc, off, 32);
    if ((threadIdx.x & 31) == 0) red[threadIdx.x >> 5] = acc;
    __syncthreads();
    if (threadIdx.x == 0) {
        float s = 0.0f;
        #pragma unroll
        for (int i = 0; i < 8; ++i) s += red[i];
        kk[n] = s;
    }
}

// ---------------------------------------------------------------------------
// Kernel 4 (main, WMMA): Kraw[i,j] = s^T w s for all 256x256 pairs.
//   Block = 32 pairs (4 i x 8 j) = two WMMA M-tiles sharing every B fragment.
//   s-tile (32x512) staged in LDS once; w streamed through LDS in 16-row
//   K-slabs (coalesced b128 stage), so the WMMA loop touches only LDS.
//   8 waves each own 4 q-tiles and keep 8 v8f accumulators across slabs.
// ---------------------------------------------------------------------------
__global__ __launch_bounds__(256) void kernel_pairK(const float* __restrict__ S,
                                                    const int* __restrict__ X1,
                                                    const int* __restrict__ X2,
                                                    const float* __restrict__ w,
                                                    float* __restrict__ Kraw) {
    __shared__ float sTile[32 * S_STRIDE];           // 66.0 KB
    __shared__ float wslab[16 * W_STRIDE];           // 33.3 KB
    __shared__ float s1s[4][S_ELE];                  //  7.1 KB
    __shared__ float s2s[8][S_ELE];                  // 14.1 KB
    __shared__ float partial[8][32];                 //  1.0 KB

    const int tid = threadIdx.x;
    const int bi = blockIdx.x >> 5;                  // 0..63  (i tiles of 4)
    const int bj = blockIdx.x & 31;                  // 0..31  (j tiles of 8)
    const int i0 = bi * 4, j0 = bj * 8;

    __builtin_prefetch(w, 0, 3);                     // global_prefetch into L2

    // Stage the per-sample 21x21 Gram blocks (gather targets) into LDS.
    for (int t = tid; t < 4 * S_ELE; t += 256) {
        int r = t / S_ELE, c = t % S_ELE;
        s1s[r][c] = S[(size_t)(i0 + r) * S_ELE + c];
    }
    for (int t = tid; t < 8 * S_ELE; t += 256) {
        int r = t / S_ELE, c = t % S_ELE;
        s2s[r][c] = S[(size_t)(NSAMP + j0 + r) * S_ELE + c];
    }
    __syncthreads();

    // Build s[m, p] = 0.5*(S1[i, a, b] + S2[j, a, b]), a=X1[i,p], b=X2[j,p].
    // X rows read coalesced straight from L2 (each reused only 4-8x here).
    for (int t = tid; t < 32 * L_SEQ; t += 256) {
        int m = t >> 9, p = t & (L_SEQ - 1);
        int mi = m >> 3, mj = m & 7;
        int a = X1[(size_t)(i0 + mi) * L_SEQ + p];
        int b = X2[(size_t)(j0 + mj) * L_SEQ + p];
        sTile[m * S_STRIDE + p] = 0.5f * (s1s[mi][a * NAA + b] + s2s[mj][a * NAA + b]);
    }

    const int wave = tid >> 5;
    const int lane = tid & 31;
    const int mrow  = lane & 15;        // A: M index / B,C,D: N index
    const int khalf = (lane >> 4) << 1; // 0 for lanes 0-15, 2 for lanes 16-31
    const int mhi   = (lane >> 4) << 3; // C/D: +8 row offset for upper lanes
    const int qbase = wave * 64;        // this wave's 4 q-tiles: qbase .. qbase+63

    v8f cacc[8] = {};                   // [qt] for m-tile0, [4+qt] for m-tile1

    for (int slab = 0; slab < 32; ++slab) {
        __syncthreads();                // previous slab fully consumed
        // Stage w[slab*16 .. +16, 0..511] -> wslab (coalesced float4).
        for (int t = tid; t < 16 * (L_SEQ / 4); t += 256) {
            int r = t >> 7, g = (t & 127) << 2;
            float4 v = *(const float4*)(w + ((size_t)(slab * 16 + r)) * L_SEQ + g);
            *(float4*)&wslab[r * W_STRIDE + g] = v;
        }
        __syncthreads();

        const int p0 = slab * 16;
        #pragma unroll
        for (int p4 = 0; p4 < 4; ++p4) {
            const int lp = p4 * 4 + khalf;          // local row in slab
            const int gp = p0 + lp;                 // global p (column of sTile)
            v2f A0, A1;
            A0.x = sTile[mrow * S_STRIDE + gp];
            A0.y = sTile[mrow * S_STRIDE + gp + 1];
            A1.x = sTile[(16 + mrow) * S_STRIDE + gp];
            A1.y = sTile[(16 + mrow) * S_STRIDE + gp + 1];
            #pragma unroll
            for (int qt = 0; qt < 4; ++qt) {
                const int q0 = qbase + qt * 16;
                v2f B;
                B.x = wslab[lp * W_STRIDE + q0 + mrow];
                B.y = wslab[(lp + 1) * W_STRIDE + q0 + mrow];
                cacc[qt]     = __builtin_amdgcn_wmma_f32_16x16x4_f32(
                                   false, A0, false, B, (short)0, cacc[qt], false, false);
                cacc[4 + qt] = __builtin_amdgcn_wmma_f32_16x16x4_f32(
                                   false, A1, false, B, (short)0, cacc[4 + qt], false, false);
            }
        }
    }

    // Epilogue: K[m] = sum_q U[m,q] * s[m,q].
    float accv[2][8] = {};
    #pragma unroll
    for (int qt = 0; qt < 4; ++qt) {
        const int q = qbase + qt * 16 + mrow;
        #pragma unroll
        for (int r = 0; r < 8; ++r) {
            accv[0][r] += cacc[qt][r]     * sTile[(r + mhi) * S_STRIDE + q];
            accv[1][r] += cacc[4 + qt][r] * sTile[(16 + r + mhi) * S_STRIDE + q];
        }
    }

    // Reduce across the 16 lanes of each half-wave; write per-wave partials.
    #pragma unroll
    for (int h = 0; h < 2; ++h) {
        #pragma unroll
        for (int r = 0; r < 8; ++r) {
            float v = accv[h][r];
            v += __shfl_xor(v, 8, 32);
            v += __shfl_xor(v, 4, 32);
            v += __shfl_xor(v, 2, 32);
            v += __shfl_xor(v, 1, 32);
            if (lane == 0)  partial[wave][h * 16 + r]     = v;   // m = 16h + r
            if (lane == 16) partial[wave][h * 16 + r + 8] = v;   // m = 16h + r + 8
        }
    }
    __syncthreads();

    if (tid < 32) {
        float s = 0.0f;
        #pragma unroll
        for (int wv = 0; wv < 8; ++wv) s += partial[wv][tid];
        int i = i0 + (tid >> 3), j = j0 + (tid & 7);
        Kraw[(size_t)i * NSAMP + j] = s;
    }
}

// ---------------------------------------------------------------------------
// Kernel 5: out[i,j] = a^2 * Kraw[i,j] / sqrt(k1[i] * k2[j])
// ---------------------------------------------------------------------------
__global__ void kernel_norm(const float* __restrict__ Kraw,
                            const float* __restrict__ kk,
                            const float* __restrict__ a,
                            float* __restrict__ out) {
    int idx = blockIdx.x * blockDim.x + threadIdx.x;
    if (idx >= NSAMP * NSAMP) return;
    int i = idx >> 8, j = idx & 255;
    float scale = a[0] * a[0];
    out[idx] = scale * Kraw[idx] * rsqrtf(kk[i]) * rsqrtf(kk[NSAMP + j]);
}

// ---------------------------------------------------------------------------
// Launch
// ---------------------------------------------------------------------------
extern "C" void kernel_launch(void* const* d_in, const int* in_sizes, int n_in,
                              void* d_out, int out_size, void* d_ws, size_t ws_size,
                              hipStream_t stream) {
    const int*   X1 = (const int*)d_in[0];      // (256, 512)
    const int*   X2 = (const int*)d_in[1];      // (256, 512)
    const float* W  = (const float*)d_in[2];    // (10752, 2688)
    const float* b  = (const float*)d_in[3];    // (2688,)
    const float* wp = (const float*)d_in[4];    // (130816,)
    const float* a  = (const float*)d_in[5];    // (1,)
    float* out = (float*)d_out;                 // (256, 256)

    // Workspace layout (floats): ~7.72 MB total
    float* w    = (float*)d_ws;                 // 512*512      = 262144
    float* E    = w + 262144;                   // 512*2688     = 1376256
    float* S    = E + 1376256;                  // 512*441      = 225792
    float* kk   = S + 225792;                   // 512
    float* Kraw = kk + 512;                     // 65536
    (void)in_sizes; (void)n_in; (void)out_size; (void)ws_size;

    kernel_buildw <<<(L_SEQ * L_SEQ + 255) / 256, 256, 0, stream>>>(wp, w);
    kernel_network<<<512, 256, 0, stream>>>(X1, X2, W, b, E);
    kernel_gram   <<<512, 256, 0, stream>>>(E, S);
    kernel_selfk  <<<512, 256, 0, stream>>>(S, X1, X2, w, kk);
    kernel_pairK  <<<2048, 256, 0, stream>>>(S, X1, X2, w, Kraw);
    kernel_norm   <<<(NSAMP * NSAMP + 255) / 256, 256, 0, stream>>>(Kraw, kk, a, out);
}